// ShortLoss_3_74689481277493
// MI455X (gfx1250) — compile-verified
//
#include <hip/hip_runtime.h>
#include <hip/hip_bf16.h>

typedef __attribute__((ext_vector_type(2))) float v2f;
typedef __attribute__((ext_vector_type(8))) float v8f;

namespace {
constexpr int kB = 512;
constexpr int kT = 4096;
constexpr int kC = 7;
constexpr int kH = 3;
constexpr int kN = kB * kT;          // 2,097,152 positions
constexpr int kThreads = 256;        // 8 wave32 waves per block
constexpr int kBlocks  = 1024;       // ~8 items per thread
constexpr int kWaves   = kThreads / 32;
}

// ---------------------------------------------------------------------------
// Wave-wide sum of 32 f32 lane partials.
// CDNA5 path: V_WMMA_F32_16X16X4_F32 with A = partials (16x4 tile = all 32
// lanes, K1/K3 zeroed) and B = ones. D[m][n] = p(m) + p(m+16), so one WMMA
// collapses the cross-half reduction; 7 adds + one shfl_xor(16) finish it.
// Exact in f32 (only additions), deterministic order.
// ---------------------------------------------------------------------------
__device__ __forceinline__ float wave_sum32(float p) {
#if __has_builtin(__builtin_amdgcn_wmma_f32_16x16x4_f32)
  v2f a; a.x = p;    a.y = 0.0f;
  v2f b; b.x = 1.0f; b.y = 1.0f;
  v8f c = {};
  v8f d = __builtin_amdgcn_wmma_f32_16x16x4_f32(
      /*neg_a=*/false, a, /*neg_b=*/false, b,
      /*c_mod=*/(short)0, c, /*reuse_a=*/false, /*reuse_b=*/false);
  float s = ((d[0] + d[1]) + (d[2] + d[3])) + ((d[4] + d[5]) + (d[6] + d[7]));
  s += __shfl_xor(s, 16, 32);
  return s;
#else
  for (int off = 16; off > 0; off >>= 1) p += __shfl_xor(p, off, 32);
  return p;
#endif
}

// Block-wide 3-way reduction. Result valid in thread 0 only.
__device__ __forceinline__ void block_reduce3(float& x, float& y, float& z,
                                              float* sm /*3*kWaves floats*/) {
  // WMMA requires EXEC all-ones: every thread of the block reaches here.
  x = wave_sum32(x);
  y = wave_sum32(y);
  z = wave_sum32(z);
  const int lane = threadIdx.x & 31;
  const int wv   = threadIdx.x >> 5;
  if (lane == 0) {
    sm[wv]              = x;
    sm[kWaves + wv]     = y;
    sm[2 * kWaves + wv] = z;
  }
  __syncthreads();
  if (threadIdx.x == 0) {
    float sx = 0.f, sy = 0.f, sz = 0.f;
#pragma unroll
    for (int i = 0; i < kWaves; ++i) {
      sx += sm[i];
      sy += sm[kWaves + i];
      sz += sm[2 * kWaves + i];
    }
    x = sx; y = sy; z = sz;
  }
}

// ---------------------------------------------------------------------------
// Pass 1: streaming per-block partials. Non-temporal loads: 235 MB single-use
// stream, don't pollute L2.
// ---------------------------------------------------------------------------
__global__ void __launch_bounds__(kThreads)
partials_kernel(const float* __restrict__ out0,
                const float* __restrict__ out1,
                const float* __restrict__ out2,
                const int*   __restrict__ labels,   // [N,3]
                const float* __restrict__ mask,     // [N]
                const float* __restrict__ reward,   // [N,3]
                float* __restrict__ ws) {           // [kBlocks,3]
  float loss_acc = 0.0f, corr_acc = 0.0f, valid_acc = 0.0f;

  const int stride = kBlocks * kThreads;
  for (int n = blockIdx.x * kThreads + threadIdx.x; n < kN; n += stride) {
    const float m = __builtin_nontemporal_load(&mask[n]);
    const bool valid = (m < 0.5f);

    float lsum = 0.0f;
    bool  allcorr = true;
    const size_t rowbase = (size_t)n * kC;
    const size_t hbase   = (size_t)n * kH;

#pragma unroll
    for (int h = 0; h < kH; ++h) {
      const float* __restrict__ row =
          (h == 0 ? out0 : (h == 1 ? out1 : out2)) + rowbase;
      const float v0 = __builtin_nontemporal_load(row + 0);
      const float v1 = __builtin_nontemporal_load(row + 1);
      const float v2 = __builtin_nontemporal_load(row + 2);
      const float v3 = __builtin_nontemporal_load(row + 3);
      const float v4 = __builtin_nontemporal_load(row + 4);
      const float v5 = __builtin_nontemporal_load(row + 5);
      const float v6 = __builtin_nontemporal_load(row + 6);

      const int   lbl = __builtin_nontemporal_load(&labels[hbase + h]);
      const float rw  = __builtin_nontemporal_load(&reward[hbase + h]);

      // argmax, first-occurrence tie-break (strict >) to match jnp.argmax
      int pred = 0; float best = v0;
      if (v1 > best) { best = v1; pred = 1; }
      if (v2 > best) { best = v2; pred = 2; }
      if (v3 > best) { best = v3; pred = 3; }
      if (v4 > best) { best = v4; pred = 4; }
      if (v5 > best) { best = v5; pred = 5; }
      if (v6 > best) { best = v6; pred = 6; }

      // gather probability of the true class (branchless selects)
      float op = v0;
      op = (lbl == 1) ? v1 : op;
      op = (lbl == 2) ? v2 : op;
      op = (lbl == 3) ? v3 : op;
      op = (lbl == 4) ? v4 : op;
      op = (lbl == 5) ? v5 : op;
      op = (lbl == 6) ? v6 : op;

      lsum += logf(op) * rw;
      allcorr = allcorr && (pred == lbl);
    }

    loss_acc  += valid ? lsum : 0.0f;
    corr_acc  += (valid && allcorr) ? 1.0f : 0.0f;
    valid_acc += valid ? 1.0f : 0.0f;
  }

  __shared__ float sm[3 * kWaves];
  block_reduce3(loss_acc, corr_acc, valid_acc, sm);
  if (threadIdx.x == 0) {
    ws[(size_t)blockIdx.x * 3 + 0] = loss_acc;
    ws[(size_t)blockIdx.x * 3 + 1] = corr_acc;
    ws[(size_t)blockIdx.x * 3 + 2] = valid_acc;
  }
}

// ---------------------------------------------------------------------------
// Pass 2: single block folds the kBlocks partials and writes the 3 outputs.
// ---------------------------------------------------------------------------
__global__ void __launch_bounds__(kThreads)
finalize_kernel(const float* __restrict__ ws, float* __restrict__ out) {
  float l = 0.0f, c = 0.0f, v = 0.0f;
  for (int i = threadIdx.x; i < kBlocks; i += kThreads) {
    l += ws[(size_t)i * 3 + 0];
    c += ws[(size_t)i * 3 + 1];
    v += ws[(size_t)i * 3 + 2];
  }
  __shared__ float sm[3 * kWaves];
  block_reduce3(l, c, v, sm);
  if (threadIdx.x == 0) {
    out[0] = -l / v;  // loss  (matches reference: div-by-zero if no valid)
    out[1] = c;       // n_correct (integer-valued, exact in f32: < 2^24)
    out[2] = v;       // n_valid
  }
}

extern "C" void kernel_launch(void* const* d_in, const int* in_sizes, int n_in,
                              void* d_out, int out_size, void* d_ws, size_t ws_size,
                              hipStream_t stream) {
  const float* out0   = (const float*)d_in[0];
  const float* out1   = (const float*)d_in[1];
  const float* out2   = (const float*)d_in[2];
  const int*   labels = (const int*)  d_in[3];  // [B,T,3] int32
  const float* mask   = (const float*)d_in[4];  // [B,T]
  const float* reward = (const float*)d_in[5];  // [B,T,3]
  float* ws  = (float*)d_ws;                    // kBlocks*3 floats (12 KB)
  float* out = (float*)d_out;                   // {loss, n_correct, n_valid}

  partials_kernel<<<kBlocks, kThreads, 0, stream>>>(out0, out1, out2,
                                                    labels, mask, reward, ws);
  finalize_kernel<<<1, kThreads, 0, stream>>>(ws, out);
}